// Network_70111046140061
// MI455X (gfx1250) — compile-verified
//
#include <hip/hip_runtime.h>
#include <hip/hip_bf16.h>
#include <stdint.h>

// ---------------------------------------------------------------------------
// Proposal generation (RetinaNet-style) for MI455X / gfx1250.
//  - top-K via exact radix-select threshold + deterministic wave32 ballot-scan
//    compaction
//  - GT table staged by the Tensor Data Mover (tensor_load_to_lds, TENSORcnt):
//    2D descriptor with stride 5 / tile width 4 strips the score column during
//    the DMA, producing a dense float4 AoS table in LDS
//  - per-block top-k index window staged via global_load_async_to_lds_b128
//    (ASYNCcnt), overlapped with the TDM transfer
//  - division-free IoU argmax (cross-multiplied comparison), LDS broadcasts
// ---------------------------------------------------------------------------

#define TOPK    10000
#define NGT_MAX 2000
#define L0      98304
#define L1      24576
#define L2      6144
#define L3      1536
#define L4      384
#define NKEEP   (TOPK + TOPK + L2 + L3 + L4)   // 28064

typedef uint32_t v4u __attribute__((ext_vector_type(4)));
typedef uint32_t v8u __attribute__((ext_vector_type(8)));

__device__ __forceinline__ unsigned int f2key(float f) {
    // monotonic float -> uint mapping (ascending)
    unsigned int u = __float_as_uint(f);
    return u ^ ((unsigned int)((int)u >> 31) | 0x80000000u);
}

// ----------------- async global -> LDS copy (CDNA5 ASYNCcnt path) ----------
__device__ __forceinline__ void async_g2l_b128(uint32_t lds_byte_addr, const void* gsrc) {
#if defined(__gfx1250__)
    // per-lane transfer: 16 bytes from the global address to the LDS address, tracked by ASYNCcnt
    asm volatile("global_load_async_to_lds_b128 %0, %1, off"
                 :: "v"(lds_byte_addr),
                    "v"((unsigned long long)(uintptr_t)gsrc)
                 : "memory");
#else
    (void)lds_byte_addr; (void)gsrc;
#endif
}

// ------------------------- Kernel 1: top-K select ---------------------------
// One 1024-thread workgroup per big level. 4-pass (8-bit digit) radix select
// finds the exact K-th largest key; then an ordered compaction emits exactly
// TOPK indices (all strictly-greater keys, plus first (K-G) ties in index
// order) using wave32 ballots + LDS wave-total scans. Fully deterministic.
__global__ __launch_bounds__(1024) void topk_select_kernel(
    const float* __restrict__ cls0, const float* __restrict__ cls1,
    int* __restrict__ sel)
{
    const int level = blockIdx.x;
    const float* __restrict__ cls = (level == 0) ? cls0 : cls1;
    const int n = (level == 0) ? L0 : L1;
    int* __restrict__ out = sel + level * TOPK;

    __shared__ unsigned int hist[256];
    __shared__ unsigned int s_bcast[2];
    __shared__ unsigned int s_wavecnt[32];

    const int tid  = threadIdx.x;
    const int lane = tid & 31;           // wave32
    const int wv   = tid >> 5;           // 32 waves / workgroup

    // safety init (d_ws is poisoned 0xAA by the harness)
    for (int t = tid; t < TOPK; t += 1024) out[t] = 0;

    // ---- radix select: find exact 32-bit key of the K-th largest element --
    unsigned int prefix = 0, prefmask = 0, want = TOPK;
    for (int shift = 24; shift >= 0; shift -= 8) {
        if (tid < 256) hist[tid] = 0;
        __syncthreads();
        for (int i = tid; i < n; i += 1024) {
            unsigned int k = f2key(cls[i]);
            if ((k & prefmask) == prefix)
                atomicAdd(&hist[(k >> shift) & 255], 1u);
        }
        __syncthreads();
        if (tid == 0) {
            unsigned int cum = 0, above = 0; int chosen = 0;
            #pragma unroll 1
            for (int b = 255; b >= 0; --b) {
                unsigned int c = hist[b];
                if (cum + c >= want) { chosen = b; above = cum; break; }
                cum += c;
                if (b == 0) { chosen = 0; above = cum; }
            }
            s_bcast[0] = (unsigned int)chosen;
            s_bcast[1] = above;              // # keys strictly greater at this digit
        }
        __syncthreads();
        unsigned int chosen = s_bcast[0];
        want   -= s_bcast[1];
        prefix |= chosen << shift;
        prefmask |= 255u << shift;
        __syncthreads();
    }
    const unsigned int thresh = prefix;      // exact K-th largest key
    // 'want' = number of keys equal to thresh we must still take (>=1)

    // ---- ordered compaction: index order, exactly TOPK outputs ------------
    unsigned int base_out = 0, base_eq = 0;
    #pragma unroll 1
    for (int start = 0; start < n; start += 1024) {
        const int i = start + tid;
        const bool valid = (i < n);
        const unsigned int k = valid ? f2key(cls[i]) : 0u;
        const bool gt = valid && (k > thresh);
        const bool eq = valid && (k == thresh);

        // scan of tie flags across the workgroup
        unsigned int beq = (unsigned int)__ballot(eq);
        unsigned int eq_in_wave = __popc(beq & ((1u << lane) - 1u));
        if (lane == 0) s_wavecnt[wv] = __popc(beq);
        __syncthreads();
        unsigned int eq_wave_base = 0, eq_total = 0;
        #pragma unroll
        for (int w = 0; w < 32; ++w) {
            unsigned int c = s_wavecnt[w];
            if (w < wv) eq_wave_base += c;
            eq_total += c;
        }
        const unsigned int eq_rank = base_eq + eq_wave_base + eq_in_wave;
        const bool take = gt || (eq && eq_rank < want);
        __syncthreads();

        // scan of selected flags
        unsigned int bsel = (unsigned int)__ballot(take);
        unsigned int sel_in_wave = __popc(bsel & ((1u << lane) - 1u));
        if (lane == 0) s_wavecnt[wv] = __popc(bsel);
        __syncthreads();
        unsigned int sel_wave_base = 0, sel_total = 0;
        #pragma unroll
        for (int w = 0; w < 32; ++w) {
            unsigned int c = s_wavecnt[w];
            if (w < wv) sel_wave_base += c;
            sel_total += c;
        }
        const unsigned int pos = base_out + sel_wave_base + sel_in_wave;
        if (take && pos < TOPK) out[pos] = i;

        base_out += sel_total;
        base_eq  += eq_total;
        __syncthreads();
    }
}

// ------------------------- Kernel 2: main compute ---------------------------
__global__ __launch_bounds__(256) void proposal_kernel(
    const float* __restrict__ anchors0, const float* __restrict__ cls0, const float* __restrict__ reg0,
    const float* __restrict__ anchors1, const float* __restrict__ cls1, const float* __restrict__ reg1,
    const float* __restrict__ anchors2, const float* __restrict__ cls2, const float* __restrict__ reg2,
    const float* __restrict__ anchors3, const float* __restrict__ cls3, const float* __restrict__ reg3,
    const float* __restrict__ anchors4, const float* __restrict__ cls4, const float* __restrict__ reg4,
    const float* __restrict__ gt_boxes, const int* __restrict__ num_gt_p,
    const int* __restrict__ sel, float* __restrict__ out)
{
    // GT table as dense float4 rows (TDM strips the score column): 32 KB
    __shared__ __align__(16) float4 gtb[NGT_MAX];
    __shared__ float garea[NGT_MAX];               // precomputed GT areas: 8 KB
    __shared__ __align__(16) int sel_sm[256];      // this block's top-k index window

    const int tid = threadIdx.x;
    const int block_base = blockIdx.x * 256;

#if defined(__gfx1250__)
    {
        // (a) ASYNCcnt path: stage this block's contiguous index window (<=1KB)
        if (block_base < 2 * TOPK) {
            const int nint = (2 * TOPK) - block_base;
            const int nchunk = (nint >= 256) ? 64 : (nint + 3) / 4;   // b128 chunks
            if (tid < nchunk)
                async_g2l_b128((uint32_t)(uintptr_t)(&sel_sm[0]) + (uint32_t)tid * 16u,
                               (const char*)(sel + block_base) + (size_t)tid * 16u);
        }
        // (b) TENSORcnt path: TDM DMA of the 2000x5 f32 GT tensor as a 2000x4
        //     tile (row stride 5 dwords, tile width 4 dwords) -> dense float4
        //     AoS in LDS. One wave issues; EXEC is ignored by tensor ops.
        if (tid < 32) {
            const unsigned long long ga = (unsigned long long)(uintptr_t)gt_boxes;
            const uint32_t lds = (uint32_t)(uintptr_t)(&gtb[0]);
            v4u g0; v8u g1;
            g0[0] = 1u;                                   // count=1, user descriptor
            g0[1] = lds;                                  // lds_addr (bytes)
            g0[2] = (uint32_t)ga;                         // global_addr[31:0]
            g0[3] = (uint32_t)(ga >> 32) | (2u << 30);    // global_addr[56:32] | type=2
            g1[0] = 2u << 16;                             // data_size = 4 bytes
            g1[1] = 5u << 16;                             // tensor_dim0 = 5 dwords/row
            g1[2] = (uint32_t)NGT_MAX << 16;              // tensor_dim1 = 2000 rows
            g1[3] = 4u << 16;                             // tile_dim0 = 4 dwords
            g1[4] = (uint32_t)NGT_MAX;                    // tile_dim1 = 2000 (tile_dim2=0)
            g1[5] = 5u;                                   // tensor_dim0_stride = 5
            g1[6] = 0u;
            g1[7] = 0u;
            asm volatile("tensor_load_to_lds %0, %1" :: "s"(g0), "s"(g1) : "memory");
#if __has_builtin(__builtin_amdgcn_s_wait_tensorcnt)
            __builtin_amdgcn_s_wait_tensorcnt(0);
#else
            asm volatile("s_wait_tensorcnt 0x0" ::: "memory");
#endif
        }
        asm volatile("s_wait_asynccnt 0" ::: "memory");
    }
#else
    if (block_base < 2 * TOPK) {
        const int idx = block_base + tid;
        if (idx < 2 * TOPK) sel_sm[tid] = sel[idx];
    }
    for (int j = tid; j < NGT_MAX; j += 256)
        gtb[j] = make_float4(gt_boxes[j * 5 + 0], gt_boxes[j * 5 + 1],
                             gt_boxes[j * 5 + 2], gt_boxes[j * 5 + 3]);
#endif
    __syncthreads();

    // precompute GT areas once per block
    for (int j = tid; j < NGT_MAX; j += 256) {
        const float4 g = gtb[j];
        garea[j] = (g.z - g.x + 1.0f) * (g.w - g.y + 1.0f);
    }
    __syncthreads();

    const int row = block_base + tid;
    if (row >= NKEEP) return;

    // map output row -> (level arrays, source index)
    const float *A, *C, *R;
    int src;
    if (row < 2 * TOPK) {
        src = sel_sm[tid];
        if (row < TOPK) {
            A = anchors0; C = cls0; R = reg0;
            src = src < 0 ? 0 : (src >= L0 ? L0 - 1 : src);
        } else {
            A = anchors1; C = cls1; R = reg1;
            src = src < 0 ? 0 : (src >= L1 ? L1 - 1 : src);
        }
    } else if (row < 2 * TOPK + L2) {
        src = row - 2 * TOPK;              A = anchors2; C = cls2; R = reg2;
    } else if (row < 2 * TOPK + L2 + L3) {
        src = row - (2 * TOPK + L2);       A = anchors3; C = cls3; R = reg3;
    } else {
        src = row - (2 * TOPK + L2 + L3);  A = anchors4; C = cls4; R = reg4;
    }

    const float4 a   = ((const float4*)A)[src];        // anchor row (16B aligned)
    const float4 dlt = ((const float4*)R)[src * 2];    // first 4 of reg8 row (32B stride)
    const float score = 1.0f / (1.0f + expf(-C[src]));

    // bbox_transform_inv
    const float w  = a.z - a.x + 1.0f, h  = a.w - a.y + 1.0f;
    const float cx = a.x + 0.5f * w,   cy = a.y + 0.5f * h;
    const float pcx = dlt.x * w + cx,  pcy = dlt.y * h + cy;
    const float pw  = expf(dlt.z) * w, ph  = expf(dlt.w) * h;
    const float px1 = pcx - 0.5f * pw, py1 = pcy - 0.5f * ph;
    const float px2 = pcx + 0.5f * pw - 1.0f, py2 = pcy + 0.5f * ph - 1.0f;
    const float area_p = (px2 - px1 + 1.0f) * (py2 - py1 + 1.0f);

    int ng = *num_gt_p;
    if (ng > NGT_MAX) ng = NGT_MAX;
    if (ng < 1) ng = 1;

    // IoU argmax, division-free: iou_j > iou_best  <=>  in_j*den_b > in_b*den_j
    // (denominators strictly positive). Strict '>' keeps the first max, like
    // jnp.argmax. All GT reads are LDS broadcasts (ds_load_b128 + ds_load_b32).
    float b_in = 0.0f, b_den = 1.0f;
    int bj = 0;
    #pragma unroll 4
    for (int j = 0; j < ng; ++j) {
        const float4 g  = gtb[j];
        const float  ag = garea[j];
        float iw = fminf(px2, g.z) - fmaxf(px1, g.x) + 1.0f;
        float ih = fminf(py2, g.w) - fmaxf(py1, g.y) + 1.0f;
        iw = fmaxf(iw, 0.0f); ih = fmaxf(ih, 0.0f);
        const float inter = iw * ih;
        const float den   = area_p + ag - inter;
        if (inter * b_den > b_in * den) { b_in = inter; b_den = den; bj = j; }
    }

    // bbox_transform(anchor, assigned gt)
    const float4 g = gtb[bj];
    const float gw  = g.z - g.x + 1.0f,  gh  = g.w - g.y + 1.0f;
    const float gcx = g.x + 0.5f * gw,   gcy = g.y + 0.5f * gh;
    const float t0 = (gcx - cx) / w;
    const float t1 = (gcy - cy) / h;
    const float t2 = logf(gw / w);
    const float t3 = logf(gh / h);

    // output row: [px1,py1,px2,py2, score, tag=1, t0,t1,t2,t3]  (8B aligned)
    float2* o2 = (float2*)(out + (size_t)row * 10);
    o2[0] = make_float2(px1, py1);
    o2[1] = make_float2(px2, py2);
    o2[2] = make_float2(score, 1.0f);
    o2[3] = make_float2(t0, t1);
    o2[4] = make_float2(t2, t3);
}

// ---------------------------------------------------------------------------
extern "C" void kernel_launch(void* const* d_in, const int* in_sizes, int n_in,
                              void* d_out, int out_size, void* d_ws, size_t ws_size,
                              hipStream_t stream)
{
    // setup_inputs() dict order: anchors0,cls0,reg0, anchors1,cls1,reg1, ...,
    // anchors4,cls4,reg4, gt_boxes, num_gt
    const float* anchors[5]; const float* cls[5]; const float* reg[5];
    for (int i = 0; i < 5; ++i) {
        anchors[i] = (const float*)d_in[3 * i + 0];
        cls[i]     = (const float*)d_in[3 * i + 1];
        reg[i]     = (const float*)d_in[3 * i + 2];
    }
    const float* gt_boxes = (const float*)d_in[15];
    const int*   num_gt   = (const int*)d_in[16];
    int*   sel = (int*)d_ws;           // 2*TOPK ints = 80 KB scratch
    float* out = (float*)d_out;        // NKEEP * 10 floats

    topk_select_kernel<<<2, 1024, 0, stream>>>(cls[0], cls[1], sel);

    const int blocks = (NKEEP + 255) / 256;
    proposal_kernel<<<blocks, 256, 0, stream>>>(
        anchors[0], cls[0], reg[0],
        anchors[1], cls[1], reg[1],
        anchors[2], cls[2], reg[2],
        anchors[3], cls[3], reg[3],
        anchors[4], cls[4], reg[4],
        gt_boxes, num_gt, sel, out);
}